// NOAM_62517543961208
// MI455X (gfx1250) — compile-verified
//
#include <hip/hip_runtime.h>
#include <math.h>

// Problem constants (from the reference)
constexpr int N = 2, C = 128, H = 96, W = 192;
constexpr int GROUPS = 8, SEARCH = 9;
constexpr int HW = H * W;                 // 18432
constexpr long CHW = (long)C * HW;

#if __has_builtin(__builtin_amdgcn_global_load_async_to_lds_b32)
#define HAVE_ASYNC_LDS 1
#else
#define HAVE_ASYNC_LDS 0
#endif

typedef __attribute__((address_space(1))) int* gptr_i32;
typedef __attribute__((address_space(3))) int* lptr_i32;

__device__ __forceinline__ void wait_async_lds() {
#if HAVE_ASYNC_LDS
#if __has_builtin(__builtin_amdgcn_s_wait_asynccnt)
  __builtin_amdgcn_s_wait_asynccnt(0);
#else
  asm volatile("s_wait_asynccnt 0" ::: "memory");
#endif
#endif
}

// ---------------------------------------------------------------------------
// Pre-pass: NCHW -> NHWC transpose of left & right features (fp32).
// 32x32 tiles through LDS; global->LDS side uses CDNA5 async-to-LDS ops
// (ASYNCcnt path) when the builtin is available.
// ---------------------------------------------------------------------------
__global__ void __launch_bounds__(256)
transpose_cx_kernel(const float* __restrict__ inL, const float* __restrict__ inR,
                    float* __restrict__ outL, float* __restrict__ outR) {
  __shared__ float tile[32][33];
  const int which = blockIdx.z >> 1;   // 0 = left, 1 = right
  const int n     = blockIdx.z & 1;    // batch (N == 2)
  const float* in = (which ? inR : inL) + (size_t)n * CHW;
  float* outp     = (which ? outR : outL) + (size_t)n * CHW;

  const int p0 = blockIdx.x * 32;      // pixel tile origin (HW dim)
  const int c0 = blockIdx.y * 32;      // channel tile origin
  const int tx = threadIdx.x;          // 0..31
  const int ty = threadIdx.y;          // 0..7

#pragma unroll
  for (int k = 0; k < 4; ++k) {
    const int r = ty + 8 * k;          // channel row within tile
    const size_t gidx = (size_t)(c0 + r) * HW + (size_t)(p0 + tx);
#if HAVE_ASYNC_LDS
    __builtin_amdgcn_global_load_async_to_lds_b32(
        (gptr_i32)(in + gidx),
        (lptr_i32)&tile[r][tx],
        /*imm offset*/ 0, /*cpol*/ 0);
#else
    tile[r][tx] = in[gidx];
#endif
  }
  wait_async_lds();
  __syncthreads();

#pragma unroll
  for (int k = 0; k < 4; ++k) {
    const int r = ty + 8 * k;          // pixel row within tile
    outp[(size_t)(p0 + r) * C + (size_t)(c0 + tx)] = tile[tx][r];
  }
}

// ---------------------------------------------------------------------------
// Main kernel: one wave32 per pixel, lanes carry 4 channels each (float4).
// NHWC=true : left/right are pre-transposed to (N,H,W,C); each bilinear tap
//             is ONE fully-coalesced b128 load per lane (512B/wave).
// NHWC=false: fallback gathering directly from NCHW (strided, correct).
// ---------------------------------------------------------------------------
template <bool NHWC>
__global__ void __launch_bounds__(256)
corr_kernel(const float* __restrict__ L, const float* __restrict__ R,
            const float* __restrict__ extra, float* __restrict__ out) {
  const int wave = blockIdx.x * (blockDim.x >> 5) + (threadIdx.x >> 5);
  const int lane = threadIdx.x & 31;

  const int n   = wave / HW;
  const int rem = wave - n * HW;       // h*W + w
  const int h   = rem / W;
  const int w   = rem - h * W;

  // Left feature for this pixel: lane holds channels 4*lane .. 4*lane+3.
  float l0, l1, l2, l3;
  if (NHWC) {
    const float4 v = reinterpret_cast<const float4*>(L + (size_t)wave * C)[lane];
    l0 = v.x; l1 = v.y; l2 = v.z; l3 = v.w;
  } else {
    const float* p = L + (size_t)n * CHW + (size_t)(4 * lane) * HW + rem;
    l0 = p[0]; l1 = p[(size_t)HW]; l2 = p[(size_t)2 * HW]; l3 = p[(size_t)3 * HW];
  }

  const float* exb = extra + (size_t)n * (2 * SEARCH) * HW + rem;

#pragma unroll
  for (int s = 0; s < SEARCH; ++s) {
    const int dx = s / 3 - 1;          // offs pattern: x-major meshgrid(ij)
    const int dy = s % 3 - 1;
    const float ex = exb[(size_t)(2 * s) * HW];
    const float ey = exb[(size_t)(2 * s + 1) * HW];
    const float fx = (float)(w + dx) + ex;
    const float fy = (float)(h + dy) + ey;
    const float x0f = floorf(fx), y0f = floorf(fy);
    const float ax = fx - x0f, ay = fy - y0f;
    const int ix = (int)x0f, iy = (int)y0f;

    float wt[4] = {(1.f - ax) * (1.f - ay),   // (x0, y0)
                   (1.f - ax) * ay,           // (x0, y1)
                   ax * (1.f - ay),           // (x1, y0)
                   ax * ay};                  // (x1, y1)
    const int xs[4] = {ix, ix, ix + 1, ix + 1};
    const int ys[4] = {iy, iy + 1, iy, iy + 1};
    int off[4];
#pragma unroll
    for (int t = 0; t < 4; ++t) {
      const bool ok = ((unsigned)xs[t] < (unsigned)W) && ((unsigned)ys[t] < (unsigned)H);
      off[t] = ok ? (ys[t] * W + xs[t]) : 0;  // zero padding == weight 0
      wt[t]  = ok ? wt[t] : 0.f;
    }

    float r0 = 0.f, r1 = 0.f, r2 = 0.f, r3 = 0.f;
#pragma unroll
    for (int t = 0; t < 4; ++t) {
      if (NHWC) {
        const float4 v = reinterpret_cast<const float4*>(
            R + ((size_t)n * HW + (size_t)off[t]) * C)[lane];
        r0 += wt[t] * v.x; r1 += wt[t] * v.y; r2 += wt[t] * v.z; r3 += wt[t] * v.w;
      } else {
        const float* p = R + (size_t)n * CHW + (size_t)(4 * lane) * HW + off[t];
        r0 += wt[t] * p[0];
        r1 += wt[t] * p[(size_t)HW];
        r2 += wt[t] * p[(size_t)2 * HW];
        r3 += wt[t] * p[(size_t)3 * HW];
      }
    }

    // Per-lane partial dot (4 channels), then reduce across the 4 lanes of
    // each group (group g == lanes 4g..4g+3 since cg = 16 = 4 lanes * 4 ch).
    float ps = l0 * r0 + l1 * r1 + l2 * r2 + l3 * r3;
    ps += __shfl_xor(ps, 1, 32);
    ps += __shfl_xor(ps, 2, 32);

    if ((lane & 3) == 0) {
      const int g = lane >> 2;
      out[((size_t)n * (GROUPS * SEARCH) + (size_t)(g * SEARCH + s)) * HW + rem] =
          ps * (1.0f / 16.0f);  // mean over cg = 16 channels
    }
  }
}

// ---------------------------------------------------------------------------
extern "C" void kernel_launch(void* const* d_in, const int* in_sizes, int n_in,
                              void* d_out, int out_size, void* d_ws, size_t ws_size,
                              hipStream_t stream) {
  const float* L = (const float*)d_in[0];
  const float* R = (const float*)d_in[1];
  const float* E = (const float*)d_in[2];
  float* out = (float*)d_out;

  const size_t elems = (size_t)N * C * H * W;
  const size_t need  = 2 * elems * sizeof(float);
  const int blocks   = (N * HW) / 8;   // 8 waves (pixels) per 256-thread block

  if (d_ws != nullptr && ws_size >= need) {
    float* LT = (float*)d_ws;
    float* RT = LT + elems;
    dim3 tb(32, 8, 1);
    dim3 tg(HW / 32, C / 32, 2 * N);   // z: (tensor, batch)
    transpose_cx_kernel<<<tg, tb, 0, stream>>>(L, R, LT, RT);
    corr_kernel<true><<<blocks, 256, 0, stream>>>(LT, RT, E, out);
  } else {
    corr_kernel<false><<<blocks, 256, 0, stream>>>(L, R, E, out);
  }
}